// ICP_54125177864548
// MI455X (gfx1250) — compile-verified
//
#include <hip/hip_runtime.h>
#include <math.h>

// ============================================================================
// ICP on MI455X (gfx1250).
// Hot loop: NN search via V_WMMA_F32_16X16X4_F32 with K=4 augmentation:
//   A(16x4, M=dst rows)  = [dx, dy, dz, -||d||^2]
//   B(4x16, N=src cols)  = [2sx, 2sy, 2sz, 1]
//   D[m][n] = 2 s.d - ||d||^2   (column-constant -||s||^2 added post-argmax)
// dst on M axis => argmax over dst is per-lane over D's 8 VGPRs.
// Argmax: 8 independent running (val, m0) chains (one per D VGPR slot).
// A-fragment LDS load software-pipelined one tile ahead (last tile peeled so
// the prefetch is unconditional -> straight-line loop body).
// dst staged to LDS via GLOBAL_LOAD_ASYNC_TO_LDS_B128 when available.
// ============================================================================

typedef __attribute__((ext_vector_type(2))) float v2f;
typedef __attribute__((ext_vector_type(8))) float v8f;
typedef __attribute__((ext_vector_type(4))) int   v4i;

#define B_ 8
#define N_ 4096
#define BN3 (B_ * 3 * N_)          // 98304
#define CH (3 * N_)                // floats per batch
#define TOL 0.001f
#define SPLIT 16                   // blocks per batch in NN kernel (128 blocks total)
#define ACC_STRIDE 16

// ---- NN kernel LDS layout (bytes) ----
#define LDS_PACK0 0                 // v2f[4096]: (dx,dy)   -> A-frag K0,K1 (lanes 0-15)
#define LDS_PACK1 (N_ * 8)          // v2f[4096]: (dz,-yy)  -> A-frag K2,K3 (lanes 16-31)
#define LDS_STAGE (2 * N_ * 8)      // float[12288]: raw async staging
#define SMEM_BYTES (2 * N_ * 8 + CH * 4)   // 112 KB

// ---- workspace float layout ----
#define WS_STATE 0   // [0]=prev_err [1]=done [2]=apply [3]=pad
#define WS_ACC   4   // 8 batches * 16 accum: [0..2]=sum s, [3..5]=sum c,
                     // [6..14]=sum s_i c_j (row-major), [15]=sum val
#define WS_RT    132 // 8 batches * 12: R row-major (9) + t (3)
#define WS_SRC   256 // mutable src copy, 98304 floats

#if __has_builtin(__builtin_amdgcn_global_load_async_to_lds_b128)
#define HAS_ASYNC_LDS 1
typedef __attribute__((address_space(1))) v4i* gv4i_p;
typedef __attribute__((address_space(3))) v4i* lv4i_p;
#else
#define HAS_ASYNC_LDS 0
#endif

// ----------------------------------------------------------------------------
__global__ void k_init(const float* __restrict__ srcInit, float* __restrict__ ws) {
    int i = blockIdx.x * blockDim.x + threadIdx.x;
    if (i < BN3) ws[WS_SRC + i] = srcInit[i];
    if (i < 4)   ws[WS_STATE + i] = 0.0f;
}

__global__ void k_zero_acc(float* __restrict__ ws) {
    int i = threadIdx.x;
    if (i < B_ * ACC_STRIDE) ws[WS_ACC + i] = 0.0f;
}

// ----------------------------------------------------------------------------
// NN search + covariance accumulation. One batch per SPLIT blocks; dst staged
// in LDS pre-packed in the ISA A-fragment layout for 16x16x4 f32 WMMA.
__global__ __launch_bounds__(256)
void k_nn(const float* __restrict__ dst, float* __restrict__ ws) {
    extern __shared__ char smem[];
    v2f* ldsA0 = (v2f*)(smem + LDS_PACK0);
    v2f* ldsA1 = (v2f*)(smem + LDS_PACK1);

    const int bid   = blockIdx.x;
    const int b     = bid / SPLIT;
    const int chunk = bid % SPLIT;
    const float* db = dst + b * CH;
    const float* sb = ws + WS_SRC + b * CH;

#if HAS_ASYNC_LDS
    // Stage raw dst (3*4096 f32 = 48KB) into LDS with async B128 copies
    // (global -> LDS, no VGPR round trip), then pack into A-fragment layout.
    {
        float* stage = (float*)(smem + LDS_STAGE);
        for (int i4 = threadIdx.x * 4; i4 < CH; i4 += 256 * 4) {
            __builtin_amdgcn_global_load_async_to_lds_b128(
                (gv4i_p)(db + i4), (lv4i_p)(stage + i4), 0, 0);
        }
#if __has_builtin(__builtin_amdgcn_s_wait_asynccnt)
        __builtin_amdgcn_s_wait_asynccnt(0);
#else
        asm volatile("s_wait_asynccnt 0x0" ::: "memory");
#endif
        __syncthreads();
        for (int i = threadIdx.x; i < N_; i += 256) {
            float dx = stage[i], dy = stage[N_ + i], dz = stage[2 * N_ + i];
            float yy = dx * dx + dy * dy + dz * dz;
            v2f a0; a0.x = dx; a0.y = dy;  ldsA0[i] = a0;
            v2f a1; a1.x = dz; a1.y = -yy; ldsA1[i] = a1;
        }
    }
#else
    for (int i = threadIdx.x; i < N_; i += 256) {
        float dx = db[i], dy = db[N_ + i], dz = db[2 * N_ + i];
        float yy = dx * dx + dy * dy + dz * dz;
        v2f a0; a0.x = dx; a0.y = dy;  ldsA0[i] = a0;
        v2f a1; a1.x = dz; a1.y = -yy; ldsA1[i] = a1;
    }
#endif
    __syncthreads();

    const int wave = threadIdx.x >> 5;
    const int lane = threadIdx.x & 31;
    const int col  = lane & 15;
    const bool hi  = lane >= 16;
    const v2f* aptr = hi ? ldsA1 : ldsA0;

    float acc[16];
#pragma unroll
    for (int q = 0; q < 16; ++q) acc[q] = 0.0f;

    const int tilesPerChunk = (N_ / 16) / SPLIT;   // 16 src tiles per block
    for (int t = wave; t < tilesPerChunk; t += 8) {
        const int n0 = (chunk * tilesPerChunk + t) * 16;
        const int n  = n0 + col;
        float sx = sb[n], sy = sb[N_ + n], sz = sb[2 * N_ + n];
        float xx = sx * sx + sy * sy + sz * sz;
        // B fragment: K0,K1 in lanes 0-15 (vgpr .x/.y), K2,K3 in lanes 16-31
        v2f bfrag;
        if (hi) { bfrag.x = 2.0f * sz; bfrag.y = 1.0f; }
        else    { bfrag.x = 2.0f * sx; bfrag.y = 2.0f * sy; }

        // 8 independent running maxima (one per D VGPR slot) -> no serial chain
        float bestv[8];
        int   bestm[8];
#pragma unroll
        for (int v = 0; v < 8; ++v) { bestv[v] = -3.0e38f; bestm[v] = 0; }

        // software pipeline, last tile peeled: prefetch is unconditional so
        // the loop body is straight-line ds_load -> wmma -> argmax
        v2f afrag = aptr[col];
        int m0 = 0;
        for (; m0 < N_ - 16; m0 += 16) {
            v2f anext = aptr[m0 + 16 + col];
            v8f c = {};
            c = __builtin_amdgcn_wmma_f32_16x16x4_f32(
                    false, afrag, false, bfrag, (short)0, c, false, false);
#pragma unroll
            for (int v = 0; v < 8; ++v) {
                float cv = c[v];
                if (cv > bestv[v]) { bestv[v] = cv; bestm[v] = m0; }
            }
            afrag = anext;
        }
        {   // last tile (m0 == N_-16)
            v8f c = {};
            c = __builtin_amdgcn_wmma_f32_16x16x4_f32(
                    false, afrag, false, bfrag, (short)0, c, false, false);
#pragma unroll
            for (int v = 0; v < 8; ++v) {
                float cv = c[v];
                if (cv > bestv[v]) { bestv[v] = cv; bestm[v] = m0; }
            }
        }
        // merge the 8 slots (branchless); prefer lower index on tie
        const int mofs = hi ? 8 : 0;
        float best = bestv[0];
        int   bidx = bestm[0] + mofs;
#pragma unroll
        for (int v = 1; v < 8; ++v) {
            int m = bestm[v] + mofs + v;
            bool take = (bestv[v] > best) | ((bestv[v] == best) & (m < bidx));
            best = take ? bestv[v] : best;
            bidx = take ? m : bidx;
        }
        // merge the two M-halves held by lane pair (n, n+16); prefer lower idx on tie
        float ob = __shfl_xor(best, 16, 32);
        int   oi = __shfl_xor(bidx, 16, 32);
        {
            bool take = (ob > best) | ((ob == best) & (oi < bidx));
            best = take ? ob : best;
            bidx = take ? oi : bidx;
        }

        v2f  c0 = ldsA0[bidx];
        float cz = ldsA1[bidx].x;
        if (!hi) {
            float val = best - xx;       // full -||s-d||^2 max
            float cx = c0.x, cy = c0.y;
            acc[0]  += sx;       acc[1]  += sy;       acc[2]  += sz;
            acc[3]  += cx;       acc[4]  += cy;       acc[5]  += cz;
            acc[6]  += sx * cx;  acc[7]  += sx * cy;  acc[8]  += sx * cz;
            acc[9]  += sy * cx;  acc[10] += sy * cy;  acc[11] += sy * cz;
            acc[12] += sz * cx;  acc[13] += sz * cy;  acc[14] += sz * cz;
            acc[15] += val;
        }
    }

#pragma unroll
    for (int q = 0; q < 16; ++q) {
        float v = acc[q];
        v += __shfl_xor(v, 1, 32);
        v += __shfl_xor(v, 2, 32);
        v += __shfl_xor(v, 4, 32);
        v += __shfl_xor(v, 8, 32);
        v += __shfl_xor(v, 16, 32);
        if (lane == 0) atomicAdd(&ws[WS_ACC + b * ACC_STRIDE + q], v);
    }
}

// ----------------------------------------------------------------------------
// Kabsch from accumulated sums: H = Sum s c^T - N sm cm^T; SVD via Jacobi on
// H^T H; R = V U^T with det-reflection fix; t = cm - R sm.
__device__ void kabsch_from_acc(const float* a, float Nf, float R[9], float t[3]) {
    float sm[3], cm[3], H[9];
#pragma unroll
    for (int i = 0; i < 3; ++i) { sm[i] = a[i] / Nf; cm[i] = a[3 + i] / Nf; }
#pragma unroll
    for (int i = 0; i < 3; ++i)
#pragma unroll
        for (int j = 0; j < 3; ++j)
            H[i * 3 + j] = a[6 + i * 3 + j] - Nf * sm[i] * cm[j];

    float K[9];
#pragma unroll
    for (int p = 0; p < 3; ++p)
#pragma unroll
        for (int q = 0; q < 3; ++q) {
            float s = 0.0f;
            for (int i = 0; i < 3; ++i) s += H[i * 3 + p] * H[i * 3 + q];
            K[p * 3 + q] = s;
        }
    float V[9] = {1, 0, 0, 0, 1, 0, 0, 0, 1};
    for (int sweep = 0; sweep < 12; ++sweep) {
        for (int pp = 0; pp < 3; ++pp) {
            int p = (pp == 2) ? 1 : 0;
            int q = (pp == 0) ? 1 : 2;
            float apq = K[p * 3 + q];
            if (fabsf(apq) < 1e-20f) continue;
            float theta = (K[q * 3 + q] - K[p * 3 + p]) / (2.0f * apq);
            float tt = ((theta >= 0.0f) ? 1.0f : -1.0f) /
                       (fabsf(theta) + sqrtf(theta * theta + 1.0f));
            float cc = 1.0f / sqrtf(tt * tt + 1.0f);
            float ss = tt * cc;
            for (int i = 0; i < 3; ++i) {
                float kip = K[i * 3 + p], kiq = K[i * 3 + q];
                K[i * 3 + p] = cc * kip - ss * kiq;
                K[i * 3 + q] = ss * kip + cc * kiq;
            }
            for (int i = 0; i < 3; ++i) {
                float kpi = K[p * 3 + i], kqi = K[q * 3 + i];
                K[p * 3 + i] = cc * kpi - ss * kqi;
                K[q * 3 + i] = ss * kpi + cc * kqi;
            }
            for (int i = 0; i < 3; ++i) {
                float vip = V[i * 3 + p], viq = V[i * 3 + q];
                V[i * 3 + p] = cc * vip - ss * viq;
                V[i * 3 + q] = ss * vip + cc * viq;
            }
        }
    }
    float w[3] = {K[0], K[4], K[8]};
    for (int i = 0; i < 2; ++i)
        for (int j = i + 1; j < 3; ++j)
            if (w[j] > w[i]) {
                float tw = w[i]; w[i] = w[j]; w[j] = tw;
                for (int r = 0; r < 3; ++r) {
                    float tv = V[r * 3 + i]; V[r * 3 + i] = V[r * 3 + j]; V[r * 3 + j] = tv;
                }
            }
    float U[9];
    for (int k = 0; k < 3; ++k) {
        float s   = sqrtf(fmaxf(w[k], 0.0f));
        float inv = 1.0f / fmaxf(s, 1e-12f);
        for (int i = 0; i < 3; ++i) {
            float hv = 0.0f;
            for (int j = 0; j < 3; ++j) hv += H[i * 3 + j] * V[j * 3 + k];
            U[i * 3 + k] = hv * inv;
        }
    }
    for (int i = 0; i < 3; ++i)
        for (int j = 0; j < 3; ++j) {
            float s = 0.0f;
            for (int k = 0; k < 3; ++k) s += V[i * 3 + k] * U[j * 3 + k];
            R[i * 3 + j] = s;
        }
    float det = R[0] * (R[4] * R[8] - R[5] * R[7])
              - R[1] * (R[3] * R[8] - R[5] * R[6])
              + R[2] * (R[3] * R[7] - R[4] * R[6]);
    if (det < 0.0f) {
        for (int r = 0; r < 3; ++r) V[r * 3 + 2] = -V[r * 3 + 2];
        for (int i = 0; i < 3; ++i)
            for (int j = 0; j < 3; ++j) {
                float s = 0.0f;
                for (int k = 0; k < 3; ++k) s += V[i * 3 + k] * U[j * 3 + k];
                R[i * 3 + j] = s;
            }
    }
    for (int i = 0; i < 3; ++i) {
        float s = 0.0f;
        for (int j = 0; j < 3; ++j) s += R[i * 3 + j] * sm[j];
        t[i] = cm[i] - s;
    }
}

__global__ void k_kabsch_step(float* __restrict__ ws) {
    int b = threadIdx.x;
    if (b < B_) {
        float R[9], t[3];
        kabsch_from_acc(&ws[WS_ACC + b * ACC_STRIDE], (float)N_, R, t);
        for (int i = 0; i < 9; ++i) ws[WS_RT + b * 12 + i] = R[i];
        for (int i = 0; i < 3; ++i) ws[WS_RT + b * 12 + 9 + i] = t[i];
    }
    if (threadIdx.x == 0) {   // scan-state update (scalar, matches jax.lax.scan)
        float sv = 0.0f;
        for (int bb = 0; bb < B_; ++bb) sv += ws[WS_ACC + bb * ACC_STRIDE + 15];
        float err  = sv / (float)(B_ * N_);
        float done = ws[WS_STATE + 1];
        float prev = ws[WS_STATE + 0];
        float apply = (done == 0.0f) ? 1.0f : 0.0f;
        float nd = (done != 0.0f || fabsf(prev - err) < TOL) ? 1.0f : 0.0f;
        if (done == 0.0f) ws[WS_STATE + 0] = err;
        ws[WS_STATE + 1] = nd;
        ws[WS_STATE + 2] = apply;
    }
}

__global__ void k_update(float* __restrict__ ws) {
    if (ws[WS_STATE + 2] == 0.0f) return;
    int i = blockIdx.x * blockDim.x + threadIdx.x;
    if (i >= B_ * N_) return;
    int b = i / N_, n = i % N_;
    float* sb = ws + WS_SRC + b * CH;
    const float* RT = ws + WS_RT + b * 12;
    float x = sb[n], y = sb[N_ + n], z = sb[2 * N_ + n];
    float nx = RT[0] * x + RT[1] * y + RT[2] * z + RT[9];
    float ny = RT[3] * x + RT[4] * y + RT[5] * z + RT[10];
    float nz = RT[6] * x + RT[7] * y + RT[8] * z + RT[11];
    sb[n] = nx; sb[N_ + n] = ny; sb[2 * N_ + n] = nz;
}

// final best-fit uses corr = final src (no NN)
__global__ void k_acc_final(const float* __restrict__ srcInit, float* __restrict__ ws) {
    int b = blockIdx.x;
    const float* s = srcInit + b * CH;
    const float* c = ws + WS_SRC + b * CH;
    float acc[15];
#pragma unroll
    for (int q = 0; q < 15; ++q) acc[q] = 0.0f;
    for (int n = threadIdx.x; n < N_; n += blockDim.x) {
        float sx = s[n], sy = s[N_ + n], sz = s[2 * N_ + n];
        float cx = c[n], cy = c[N_ + n], cz = c[2 * N_ + n];
        acc[0]  += sx;       acc[1]  += sy;       acc[2]  += sz;
        acc[3]  += cx;       acc[4]  += cy;       acc[5]  += cz;
        acc[6]  += sx * cx;  acc[7]  += sx * cy;  acc[8]  += sx * cz;
        acc[9]  += sy * cx;  acc[10] += sy * cy;  acc[11] += sy * cz;
        acc[12] += sz * cx;  acc[13] += sz * cy;  acc[14] += sz * cz;
    }
    int lane = threadIdx.x & 31;
#pragma unroll
    for (int q = 0; q < 15; ++q) {
        float v = acc[q];
        v += __shfl_xor(v, 1, 32);
        v += __shfl_xor(v, 2, 32);
        v += __shfl_xor(v, 4, 32);
        v += __shfl_xor(v, 8, 32);
        v += __shfl_xor(v, 16, 32);
        if (lane == 0) atomicAdd(&ws[WS_ACC + b * ACC_STRIDE + q], v);
    }
}

__global__ void k_kabsch_final(float* __restrict__ ws, float* __restrict__ out) {
    int b = threadIdx.x;
    if (b >= B_) return;
    float R[9], t[3];
    kabsch_from_acc(&ws[WS_ACC + b * ACC_STRIDE], (float)N_, R, t);
    float* rab = out + 2 * BN3;        // 196608
    float* tab = rab + B_ * 9;         // 196680
    float* rba = tab + B_ * 3;         // 196704
    float* tba = rba + B_ * 9;         // 196776
    for (int i = 0; i < 3; ++i)
        for (int j = 0; j < 3; ++j) {
            rab[b * 9 + i * 3 + j] = R[i * 3 + j];
            rba[b * 9 + i * 3 + j] = R[j * 3 + i];
        }
    for (int i = 0; i < 3; ++i) tab[b * 3 + i] = t[i];
    for (int i = 0; i < 3; ++i) {
        float s = 0.0f;
        for (int j = 0; j < 3; ++j) s += R[j * 3 + i] * t[j];
        tba[b * 3 + i] = -s;
    }
}

__global__ void k_copy_out(const float* __restrict__ srcInit,
                           const float* __restrict__ ws, float* __restrict__ out) {
    int i = blockIdx.x * blockDim.x + threadIdx.x;
    if (i < BN3) { out[i] = srcInit[i]; out[BN3 + i] = ws[WS_SRC + i]; }
}

// ----------------------------------------------------------------------------
extern "C" void kernel_launch(void* const* d_in, const int* in_sizes, int n_in,
                              void* d_out, int out_size, void* d_ws, size_t ws_size,
                              hipStream_t stream) {
    const float* srcInit = (const float*)d_in[0];
    const float* dst     = (const float*)d_in[1];
    float* out = (float*)d_out;
    float* ws  = (float*)d_ws;

    k_init<<<(BN3 + 255) / 256, 256, 0, stream>>>(srcInit, ws);
    for (int it = 0; it < 10; ++it) {
        k_zero_acc<<<1, 128, 0, stream>>>(ws);
        k_nn<<<B_ * SPLIT, 256, SMEM_BYTES, stream>>>(dst, ws);
        k_kabsch_step<<<1, 32, 0, stream>>>(ws);
        k_update<<<(B_ * N_ + 255) / 256, 256, 0, stream>>>(ws);
    }
    k_zero_acc<<<1, 128, 0, stream>>>(ws);
    k_acc_final<<<B_, 256, 0, stream>>>(srcInit, ws);
    k_kabsch_final<<<1, 32, 0, stream>>>(ws, out);
    k_copy_out<<<(BN3 + 255) / 256, 256, 0, stream>>>(srcInit, ws, out);
}